// Attention_Distance_61873298866224
// MI455X (gfx1250) — compile-verified
//
#include <hip/hip_runtime.h>
#include <math.h>

typedef __attribute__((ext_vector_type(2))) float v2f;
typedef __attribute__((ext_vector_type(4))) float v4f;
typedef __attribute__((ext_vector_type(8))) float v8f;

#define N_FEAT  64
#define N_HEADS 4

// ---- monotonic float <-> uint key (for atomic max over floats) ----
__device__ __forceinline__ unsigned f2key(float f) {
  unsigned u = __float_as_uint(f);
  return (u & 0x80000000u) ? ~u : (u | 0x80000000u);
}
__device__ __forceinline__ float key2f(unsigned k) {
  unsigned u = (k & 0x80000000u) ? (k & 0x7FFFFFFFu) : ~k;
  return __uint_as_float(u);
}

// ws layout (floats): [0..3] per-head sums, [4] max key (as uint),
//                     [8 .. 8+N*4) per-(node,head) segment sums
#define WS_SEG_OFF 8

// ---------------- K0: init accumulators ----------------
__global__ void k0_init(float* __restrict__ ws, int segCount) {
  int i = blockIdx.x * blockDim.x + threadIdx.x;
  if (i == 0) {
    ws[0] = 0.f; ws[1] = 0.f; ws[2] = 0.f; ws[3] = 0.f;
    ((unsigned*)ws)[4] = 0x007FFFFFu;   // f2key(-inf)
  }
  if (i < segCount) ws[WS_SEG_OFF + i] = 0.f;
}

// ---------------- K1: per-edge distances via V_WMMA_F32_16X16X4_F32 ----------------
// One wave32 handles a tile of 16 edges. D[16x16] = A[16x64] x B[64x16],
// K consumed 4 at a time (16 WMMA steps). A[m,k] = (nf[trg]-nf[src])^2,
// B[k,n] = ew[n,k]^2 * dw[n,k] for heads n<4, 0 otherwise.
// Tail lanes clamp to edge E-1 and duplicate-write identical data (benign),
// so the hot loop is branch-free and EXEC stays all-ones for WMMA.
__global__ void k1_edge_dist(const float* __restrict__ nodes,
                             const long long* __restrict__ eidx,
                             const float* __restrict__ ew,
                             const float* __restrict__ dw,
                             float* __restrict__ out,   // [0,E*4): ed scratch; [E*4,..): nf_src
                             float* __restrict__ ws,
                             long long E) {
  __shared__ float lds_sum[N_HEADS];
  if (threadIdx.x < N_HEADS) lds_sum[threadIdx.x] = 0.f;
  __syncthreads();

  const int lane = (int)(threadIdx.x & 31u);
  const int wv   = (int)(threadIdx.x >> 5);
  const int m    = lane & 15;      // edge-in-tile (A row) / head col (B)
  const int half = lane >> 4;      // K-half selector within each 4-wide chunk
  const long long base = ((long long)blockIdx.x * 8 + wv) * 16;

  const long long e  = base + m;
  const long long ec = (e < E) ? e : (E - 1);   // clamp tail lanes
  const long long trg = eidx[ec];
  const long long src = eidx[E + ec];
  // Each lane covers features { 4s + 2*half, 4s + 2*half + 1 }, s = 0..15
  const float* __restrict__ ft = nodes + trg * N_FEAT + 2 * half;
  const float* __restrict__ fs = nodes + src * N_FEAT + 2 * half;
  float* __restrict__ nsrc = out + E * (long long)N_HEADS + ec * N_FEAT + 2 * half;

  // Preload B operand: lane covers head column n = m; columns >= 4 are zero.
  v2f B[16];
#pragma unroll
  for (int s = 0; s < 16; ++s) {
    const int k0 = 4 * s + 2 * half;
    v2f b; b.x = 0.f; b.y = 0.f;
    if (m < N_HEADS) {
      float w0 = ew[m * N_FEAT + k0],     d0 = dw[m * N_FEAT + k0];
      float w1 = ew[m * N_FEAT + k0 + 1], d1 = dw[m * N_FEAT + k0 + 1];
      b.x = w0 * w0 * d0;
      b.y = w1 * w1 * d1;
    }
    B[s] = b;
  }

  // Batch all 32 gathers up front: 32 outstanding b64 loads hide L2 latency.
  v2f t[16], sv[16];
#pragma unroll
  for (int s = 0; s < 16; ++s) {
    t[s]  = *(const v2f*)(ft + 4 * s);
    sv[s] = *(const v2f*)(fs + 4 * s);
  }

  v8f acc = {};
#pragma unroll
  for (int s = 0; s < 16; ++s) {
    *(v2f*)(nsrc + 4 * s) = sv[s];   // stream nf_src out (unconditional, clamped)
    v2f a = t[s] - sv[s];
    a *= a;
    // D = A*B + C  (f32, 16x16x4)
    acc = __builtin_amdgcn_wmma_f32_16x16x4_f32(false, a, false, B[s],
                                                (short)0, acc, false, false);
  }

  // D layout: lane holds column n=m, rows M = 8*half + v  (v = VGPR index)
  float lsum = 0.f;
  if (m < N_HEADS) {
#pragma unroll
    for (int v = 0; v < 8; ++v) {
      long long eM = base + 8 * half + v;
      if (eM < E) {
        float d = acc[v];
        out[eM * N_HEADS + m] = d;   // ed scratch in att region
        lsum += d;
      }
    }
    atomicAdd(&lds_sum[m], lsum);
  }
  __syncthreads();
  if (threadIdx.x < N_HEADS) atomicAdd(&ws[threadIdx.x], lds_sum[threadIdx.x]);
}

// ---------------- K2: scores = -leaky_relu(ed + mean); global max ----------------
__global__ void k2_score_max(float* __restrict__ out, float* __restrict__ ws, long long E) {
  __shared__ unsigned lmax;
  if (threadIdx.x == 0) lmax = 0u;
  __syncthreads();

  const long long e = (long long)blockIdx.x * blockDim.x + threadIdx.x;
  unsigned k = 0u;
  if (e < E) {
    v4f ed = *(v4f*)(out + e * 4);
    const float invE = 1.f / (float)E;
    v4f sc;
#pragma unroll
    for (int h = 0; h < N_HEADS; ++h) {
      float x = ed[h] + ws[h] * invE;
      float l = (x > 0.f) ? x : 0.2f * x;
      float s = -l;
      sc[h] = s;
      unsigned kk = f2key(s);
      k = (kk > k) ? kk : k;
    }
    *(v4f*)(out + e * 4) = sc;
  }
  atomicMax(&lmax, k);
  __syncthreads();
  if (threadIdx.x == 0) atomicMax(((unsigned*)ws) + 4, lmax);
}

// ---------------- K3: p = exp(score - max); segment sums via atomics ----------------
__global__ void k3_exp_segsum(float* __restrict__ out,
                              const long long* __restrict__ eidx,
                              float* __restrict__ ws, long long E) {
  const long long e = (long long)blockIdx.x * blockDim.x + threadIdx.x;
  if (e >= E) return;
  const float mx = key2f(((const unsigned*)ws)[4]);
  v4f sc = *(v4f*)(out + e * 4);
  const long long trg = eidx[e];
  float* __restrict__ seg = ws + WS_SEG_OFF + trg * N_HEADS;
  v4f p;
#pragma unroll
  for (int h = 0; h < N_HEADS; ++h) {
    float v = __expf(sc[h] - mx);
    p[h] = v;
    atomicAdd(&seg[h], v);
  }
  *(v4f*)(out + e * 4) = p;
}

// ---------------- K4: att = p / (segsum[trg] + 1e-16) ----------------
__global__ void k4_normalize(float* __restrict__ out,
                             const long long* __restrict__ eidx,
                             const float* __restrict__ ws, long long E) {
  const long long e = (long long)blockIdx.x * blockDim.x + threadIdx.x;
  if (e >= E) return;
  const long long trg = eidx[e];
  v4f p = *(v4f*)(out + e * 4);
  v4f d = *(const v4f*)(ws + WS_SEG_OFF + trg * N_HEADS);
  v4f a;
#pragma unroll
  for (int h = 0; h < N_HEADS; ++h) a[h] = p[h] / (d[h] + 1e-16f);
  *(v4f*)(out + e * 4) = a;
}

extern "C" void kernel_launch(void* const* d_in, const int* in_sizes, int n_in,
                              void* d_out, int out_size, void* d_ws, size_t ws_size,
                              hipStream_t stream) {
  const float*     nodes = (const float*)d_in[0];
  const long long* eidx  = (const long long*)d_in[1];
  const float*     ew    = (const float*)d_in[2];
  const float*     dw    = (const float*)d_in[3];
  float* out = (float*)d_out;
  float* ws  = (float*)d_ws;

  const long long N = in_sizes[0] / N_FEAT;
  const long long E = in_sizes[1] / 2;
  const int segCount = (int)(N * N_HEADS);

  // K0: init accumulators (must re-init every call: graph-replay safe)
  k0_init<<<(unsigned)((segCount + 255) / 256), 256, 0, stream>>>(ws, segCount);

  // K1: 16 edges per wave32, 8 waves per block -> 128 edges / block
  const long long blocks1 = (E + 127) / 128;
  k1_edge_dist<<<(unsigned)blocks1, 256, 0, stream>>>(nodes, eidx, ew, dw, out, ws, E);

  const unsigned blocksE = (unsigned)((E + 255) / 256);
  k2_score_max <<<blocksE, 256, 0, stream>>>(out, ws, E);
  k3_exp_segsum<<<blocksE, 256, 0, stream>>>(out, eidx, ws, E);
  k4_normalize <<<blocksE, 256, 0, stream>>>(out, eidx, ws, E);
}